// WassersteinDistGS_11897059410345
// MI455X (gfx1250) — compile-verified
//
#include <hip/hip_runtime.h>
#include <math.h>

#ifndef __has_builtin
#define __has_builtin(x) 0
#endif

#define TILE 256

// int4 vector type matching the async-to-LDS builtin's expected pointee type,
// plus address-space-qualified pointee typedefs (AS1 = global, AS3 = LDS).
typedef int v4i __attribute__((vector_size(16)));
typedef __attribute__((address_space(1))) v4i global_v4i;
typedef __attribute__((address_space(3))) v4i local_v4i;

// ---------------------------------------------------------------------------
// Cooperative tile staging: global -> LDS in 128-bit chunks.
// Prefer the gfx1250 async-to-LDS path (ASYNCcnt-tracked loads that bypass
// VGPRs); fall back to load+ds_store if the builtin is unavailable.
// ---------------------------------------------------------------------------
__device__ __forceinline__ void stage_copy(float* lds_dst, const float* g_src,
                                           int nfloats) {
  const int n4 = nfloats >> 2;  // callers pass multiples of 4
  const float4* g = (const float4*)g_src;
  float4* l = (float4*)lds_dst;
  for (int i = threadIdx.x; i < n4; i += TILE) {
#if __has_builtin(__builtin_amdgcn_global_load_async_to_lds_b128)
    __builtin_amdgcn_global_load_async_to_lds_b128(
        (global_v4i*)(g + i), (local_v4i*)(l + i),
        /*offset=*/0, /*cpol=*/0);
#else
    l[i] = g[i];
#endif
  }
}

__device__ __forceinline__ void wait_async_lds() {
#if __has_builtin(__builtin_amdgcn_global_load_async_to_lds_b128)
#if __has_builtin(__builtin_amdgcn_s_wait_asynccnt)
  __builtin_amdgcn_s_wait_asynccnt(0);
#else
  asm volatile("s_wait_asynccnt 0" ::: "memory");
#endif
#endif
}

// ---------------------------------------------------------------------------
// Per-element 2-Wasserstein distance between two 3D Gaussians.
//   cov = R diag(s) R^T ;  W^2 = |mu0-mu1|^2 + tr(c0) + tr(c1) - 2 tr(sqrt(E))
// with E = D M S1 M^T D, M = R0^T R1, D = diag(sqrt(s0)).
// E = N N^T with N = D M diag(sqrt(s1))  (symmetric PSD by construction).
// Eigenvalues of the symmetric 3x3 E via Smith's closed form.
// ---------------------------------------------------------------------------
__device__ __forceinline__ float wdist_elem(
    const float* __restrict__ l0, const float* __restrict__ s0,
    const float* __restrict__ R0, const float* __restrict__ l1,
    const float* __restrict__ s1, const float* __restrict__ R1) {
  const float dx = l0[0] - l1[0], dy = l0[1] - l1[1], dz = l0[2] - l1[2];
  const float loc_diff2 = dx * dx + dy * dy + dz * dz;

  // M = R0^T * R1   (row-major 3x3)
  float M[9];
#pragma unroll
  for (int i = 0; i < 3; ++i)
#pragma unroll
    for (int k = 0; k < 3; ++k)
      M[i * 3 + k] =
          R0[i] * R1[k] + R0[3 + i] * R1[3 + k] + R0[6 + i] * R1[6 + k];

  const float sa0 = sqrtf(s0[0]), sa1 = sqrtf(s0[1]), sa2 = sqrtf(s0[2]);
  const float sb0 = sqrtf(s1[0]), sb1 = sqrtf(s1[1]), sb2 = sqrtf(s1[2]);

  float N[9];
  N[0] = sa0 * M[0] * sb0; N[1] = sa0 * M[1] * sb1; N[2] = sa0 * M[2] * sb2;
  N[3] = sa1 * M[3] * sb0; N[4] = sa1 * M[4] * sb1; N[5] = sa1 * M[5] * sb2;
  N[6] = sa2 * M[6] * sb0; N[7] = sa2 * M[7] * sb1; N[8] = sa2 * M[8] * sb2;

  // E = N N^T (6 unique entries)
  const float e00 = N[0] * N[0] + N[1] * N[1] + N[2] * N[2];
  const float e11 = N[3] * N[3] + N[4] * N[4] + N[5] * N[5];
  const float e22 = N[6] * N[6] + N[7] * N[7] + N[8] * N[8];
  const float e01 = N[0] * N[3] + N[1] * N[4] + N[2] * N[5];
  const float e02 = N[0] * N[6] + N[1] * N[7] + N[2] * N[8];
  const float e12 = N[3] * N[6] + N[4] * N[7] + N[5] * N[8];

  const float cov0_tr = s0[0] + s0[1] + s0[2];
  const float cov1_tr =
      s1[0] * (R1[0] * R1[0] + R1[3] * R1[3] + R1[6] * R1[6]) +
      s1[1] * (R1[1] * R1[1] + R1[4] * R1[4] + R1[7] * R1[7]) +
      s1[2] * (R1[2] * R1[2] + R1[5] * R1[5] + R1[8] * R1[8]);

  // Closed-form eigenvalues of symmetric E (branchless guard for p ~ 0).
  const float q = (e00 + e11 + e22) * (1.0f / 3.0f);
  const float a0 = e00 - q, a1 = e11 - q, a2 = e22 - q;
  const float p2 = a0 * a0 + a1 * a1 + a2 * a2 +
                   2.0f * (e01 * e01 + e02 * e02 + e12 * e12);
  const float p = sqrtf(p2 * (1.0f / 6.0f));
  const float ip = 1.0f / fmaxf(p, 1e-30f);
  const float b00 = a0 * ip, b11 = a1 * ip, b22 = a2 * ip;
  const float b01 = e01 * ip, b02 = e02 * ip, b12 = e12 * ip;
  float r = 0.5f * (b00 * (b11 * b22 - b12 * b12) -
                    b01 * (b01 * b22 - b02 * b12) +
                    b02 * (b01 * b12 - b02 * b11));
  r = fminf(1.0f, fmaxf(-1.0f, r));
  const float phi = acosf(r) * (1.0f / 3.0f);
  const float twop = 2.0f * p;
  const float lam1 = q + twop * __cosf(phi);
  const float lam3 = q + twop * __cosf(phi + 2.0943951023931953f);  // +2pi/3
  const float lam2 = 3.0f * q - lam1 - lam3;

  // tr(sqrt(E)) = sum |lambda|^1/2  (reference: (l^2)^(1/4))
  const float est = sqrtf(fabsf(lam1)) + sqrtf(fabsf(lam2)) + sqrtf(fabsf(lam3));
  const float cov_w = fmaxf(cov0_tr + cov1_tr - 2.0f * est, 1e-8f);
  return sqrtf(loc_diff2 + cov_w);
}

__global__ __launch_bounds__(TILE) void WassersteinDistGS_kernel(
    const float* __restrict__ loc0, const float* __restrict__ scale0,
    const float* __restrict__ rot0, const float* __restrict__ loc1,
    const float* __restrict__ scale1, const float* __restrict__ rot1,
    float* __restrict__ out, int n) {
  __shared__ float smem[TILE * 30];  // 30 KB: all six arrays for one tile
  float* s_loc0 = smem;               // TILE*3
  float* s_s0   = smem + TILE * 3;    // TILE*3
  float* s_rot0 = smem + TILE * 6;    // TILE*9
  float* s_loc1 = smem + TILE * 15;   // TILE*3
  float* s_s1   = smem + TILE * 18;   // TILE*3
  float* s_rot1 = smem + TILE * 21;   // TILE*9

  const int base = blockIdx.x * TILE;
  const int t = threadIdx.x;
  const int gid = base + t;

  if (base + TILE <= n) {  // block-uniform branch
    // Fully coalesced 128-bit staging into LDS (async path on gfx1250).
    stage_copy(s_loc0, loc0 + base * 3, TILE * 3);
    stage_copy(s_s0,   scale0 + base * 3, TILE * 3);
    stage_copy(s_rot0, rot0 + base * 9, TILE * 9);
    stage_copy(s_loc1, loc1 + base * 3, TILE * 3);
    stage_copy(s_s1,   scale1 + base * 3, TILE * 3);
    stage_copy(s_rot1, rot1 + base * 9, TILE * 9);
    wait_async_lds();
    __syncthreads();
    // LDS reads: comp strides 3 and 9 are coprime with 64 banks -> conflict-free.
    out[gid] = wdist_elem(s_loc0 + t * 3, s_s0 + t * 3, s_rot0 + t * 9,
                          s_loc1 + t * 3, s_s1 + t * 3, s_rot1 + t * 9);
  } else if (gid < n) {  // generic tail (not hit for B = 4096*256)
    float a[3], b[3], RA[9], c[3], d[3], RB[9];
#pragma unroll
    for (int k = 0; k < 3; ++k) {
      a[k] = loc0[gid * 3 + k];
      b[k] = scale0[gid * 3 + k];
      c[k] = loc1[gid * 3 + k];
      d[k] = scale1[gid * 3 + k];
    }
#pragma unroll
    for (int k = 0; k < 9; ++k) {
      RA[k] = rot0[gid * 9 + k];
      RB[k] = rot1[gid * 9 + k];
    }
    out[gid] = wdist_elem(a, b, RA, c, d, RB);
  }
}

extern "C" void kernel_launch(void* const* d_in, const int* in_sizes, int n_in,
                              void* d_out, int out_size, void* d_ws,
                              size_t ws_size, hipStream_t stream) {
  (void)n_in; (void)d_ws; (void)ws_size; (void)out_size;
  const float* loc0   = (const float*)d_in[0];
  const float* scale0 = (const float*)d_in[1];
  const float* rot0   = (const float*)d_in[2];
  const float* loc1   = (const float*)d_in[3];
  const float* scale1 = (const float*)d_in[4];
  const float* rot1   = (const float*)d_in[5];
  float* out = (float*)d_out;

  const int n = in_sizes[0] / 3;  // loc0 is [B,3]
  const int grid = (n + TILE - 1) / TILE;
  WassersteinDistGS_kernel<<<grid, TILE, 0, stream>>>(
      loc0, scale0, rot0, loc1, scale1, rot1, out, n);
}